// SAGE_36240934043672
// MI455X (gfx1250) — compile-verified
//
#include <hip/hip_runtime.h>

#define IN_CH 128
#define HID   256

typedef __attribute__((ext_vector_type(2))) float v2f;
typedef __attribute__((ext_vector_type(8))) float v8f;

// ---------------- degree / aggregation helpers ----------------

__global__ void deg_kernel(const int* __restrict__ dst, float* __restrict__ deg, int E) {
    int e = blockIdx.x * blockDim.x + threadIdx.x;
    if (e < E) atomicAdd(&deg[dst[e]], 1.0f);
}

__global__ void deginv_kernel(float* __restrict__ deg, int n) {
    int i = blockIdx.x * blockDim.x + threadIdx.x;
    if (i < n) deg[i] = 1.0f / fmaxf(deg[i], 1.0f);
}

// one thread per (edge, 4-channel group); channels = 1<<cshift.
// float4 gather (16B aligned: row starts are 512B/1KB aligned, cq*16B offsets)
// followed by 4 coalesced f32 atomics into the destination row.
__global__ void scatter_kernel4(const float* __restrict__ h,
                                const int* __restrict__ src,
                                const int* __restrict__ dst,
                                const float* __restrict__ ew,
                                float* __restrict__ agg,
                                long long total4, int cshift) {
    long long i = (long long)blockIdx.x * blockDim.x + threadIdx.x;
    if (i >= total4) return;
    int qshift = cshift - 2;                       // channel-quads per row
    int cq = (int)(i & ((1 << qshift) - 1));
    int e  = (int)(i >> qshift);
    int s = src[e];
    int d = dst[e];
    float w = ew[e];
    const float4* hp = (const float4*)(h + ((long long)s << cshift));
    float4 hv = hp[cq];
    float* ap = agg + ((long long)d << cshift) + ((long long)cq << 2);
    atomicAdd(ap + 0, hv.x * w);
    atomicAdd(ap + 1, hv.y * w);
    atomicAdd(ap + 2, hv.z * w);
    atomicAdd(ap + 3, hv.w * w);
}

__global__ void scale_kernel4(float4* __restrict__ agg, const float* __restrict__ deginv,
                              long long total4, int qshift) {
    long long i = (long long)blockIdx.x * blockDim.x + threadIdx.x;
    if (i >= total4) return;
    float s = deginv[i >> qshift];
    float4 v = agg[i];
    v.x *= s; v.y *= s; v.z *= s; v.w *= s;
    agg[i] = v;
}

// ---------------- fused dual-GEMM with f32 WMMA ----------------
// out[M,N] = Aagg[M,K] @ Bl[K,N] + Aself[M,K] @ Br[K,N] + bias[N]  (opt. ReLU)
// One wave computes a (16*MT) x 64 output slab with V_WMMA_F32_16X16X4_F32.
// Per k-step: 2 x b64 A-loads + 8 x b32 B-loads feed 8 back-to-back WMMAs.
// M-tail (odd m-tile count) is handled by duplicating the last m-tile in
// compute and skipping its store — all guards are wave-uniform, so EXEC is
// all-ones at every WMMA as the ISA requires.
__launch_bounds__(256)
__global__ void sage_gemm_wmma(const float* __restrict__ Aagg,
                               const float* __restrict__ Bl,
                               const float* __restrict__ Aself,
                               const float* __restrict__ Br,
                               const float* __restrict__ bias,
                               float* __restrict__ out,
                               int M, int K, int N, int relu) {
    const int MT = 2;                       // m-tiles (16 rows each) per wave
    const int NT = 4;                       // n-tiles (16 cols each) per wave
    int lane = threadIdx.x & 31;
    int wave = (int)((blockIdx.x * (long long)blockDim.x + threadIdx.x) >> 5);
    int nGroups = N / (16 * NT);
    int mTiles  = M / 16;
    int mGroups = (mTiles + MT - 1) / MT;
    int jobs = mGroups * nGroups;
    if (wave >= jobs) return;               // whole wave exits together

    int mg = wave / nGroups;
    int nBase = (wave % nGroups) * (16 * NT);
    int mTile0 = mg * MT;
    int mCount = (mTiles - mTile0 < MT) ? (mTiles - mTile0) : MT;  // uniform

    int half = lane >> 4;                   // 0: lanes 0-15, 1: lanes 16-31
    int ml   = lane & 15;
    int kOff = half * 2;                    // upper half-wave takes K+2,K+3

    // A rows for this lane; tail wave duplicates row of m-tile 0
    int arow[MT];
    arow[0] = mTile0 * 16 + ml;
    arow[1] = (mCount > 1) ? (mTile0 + 1) * 16 + ml : arow[0];

    v8f acc[MT][NT] = {};

    #pragma unroll 1
    for (int term = 0; term < 2; ++term) {
        const float* A = term ? Aself : Aagg;
        const float* B = term ? Br : Bl;
        const float* Ar0 = A + (long long)arow[0] * K;
        const float* Ar1 = A + (long long)arow[1] * K;
        for (int k0 = 0; k0 < K; k0 += 4) {
            // ---- gather all operands for this k-step first ----
            v2f a[MT];
            a[0].x = Ar0[k0 + kOff];
            a[0].y = Ar0[k0 + kOff + 1];
            a[1].x = Ar1[k0 + kOff];
            a[1].y = Ar1[k0 + kOff + 1];
            const float* Bp = B + (long long)(k0 + kOff) * N + nBase + ml;
            v2f b[NT];
            #pragma unroll
            for (int t = 0; t < NT; ++t) {
                b[t].x = Bp[t * 16];        // row k0+kOff
                b[t].y = Bp[N + t * 16];    // row k0+kOff+1
            }
            // ---- then burst the WMMAs (independent accumulators) ----
            #pragma unroll
            for (int t = 0; t < NT; ++t) {
                #pragma unroll
                for (int m = 0; m < MT; ++m) {
                    acc[m][t] = __builtin_amdgcn_wmma_f32_16x16x4_f32(
                        false, a[m], false, b[t],
                        (short)0, acc[m][t], false, false);
                }
            }
        }
    }

    // C/D layout: VGPR r -> row (r + 8*half), col = ml (within tile)
    #pragma unroll
    for (int m = 0; m < MT; ++m) {
        if (m < mCount) {                   // uniform guard
            #pragma unroll
            for (int t = 0; t < NT; ++t) {
                int col = nBase + t * 16 + ml;
                float bv = bias[col];
                #pragma unroll
                for (int r = 0; r < 8; ++r) {
                    int row = (mTile0 + m) * 16 + half * 8 + r;
                    float v = acc[m][t][r] + bv;
                    if (relu) v = fmaxf(v, 0.0f);
                    out[(long long)row * N + col] = v;
                }
            }
        }
    }
}

// ---------------- host-side launch ----------------

extern "C" void kernel_launch(void* const* d_in, const int* in_sizes, int n_in,
                              void* d_out, int out_size, void* d_ws, size_t ws_size,
                              hipStream_t stream) {
    const float* x   = (const float*)d_in[0];
    const int*   ei  = (const int*)  d_in[1];   // [2, E]: src row then dst row
    const float* ew  = (const float*)d_in[2];
    const float* Wl0 = (const float*)d_in[3];
    const float* Wr0 = (const float*)d_in[4];
    const float* b0  = (const float*)d_in[5];
    const float* Wl1 = (const float*)d_in[6];
    const float* Wr1 = (const float*)d_in[7];
    const float* b1  = (const float*)d_in[8];

    const int E = in_sizes[2];
    const int M = in_sizes[0] / IN_CH;          // 50000, multiple of 16

    const int* src = ei;
    const int* dst = ei + E;

    // workspace layout (floats)
    float* ws   = (float*)d_ws;
    float* deg  = ws;                                   // M
    float* agg0 = deg  + M;                             // M * IN_CH
    float* h0   = agg0 + (size_t)M * IN_CH;             // M * HID
    float* agg1 = h0   + (size_t)M * HID;               // M * HID
    float* out  = (float*)d_out;

    hipMemsetAsync(deg,  0, (size_t)M * sizeof(float), stream);
    hipMemsetAsync(agg0, 0, (size_t)M * IN_CH * sizeof(float), stream);
    hipMemsetAsync(agg1, 0, (size_t)M * HID * sizeof(float), stream);

    // in-degree and reciprocal
    deg_kernel<<<(E + 255) / 256, 256, 0, stream>>>(dst, deg, E);
    deginv_kernel<<<(M + 255) / 256, 256, 0, stream>>>(deg, M);

    // ---- layer 0 ----
    long long tot0 = (long long)E * (IN_CH / 4);
    scatter_kernel4<<<(unsigned)((tot0 + 255) / 256), 256, 0, stream>>>(
        x, src, dst, ew, agg0, tot0, 7 /* log2(128) */);
    long long sc0 = (long long)M * (IN_CH / 4);
    scale_kernel4<<<(unsigned)((sc0 + 255) / 256), 256, 0, stream>>>(
        (float4*)agg0, deg, sc0, 5 /* log2(128/4) */);

    int mGroups = (M / 16 + 1) / 2;            // ceil(3125/2) = 1563
    int jobs = mGroups * (HID / 64);           // 1563 * 4 = 6252 waves
    int gthreads = jobs * 32;
    sage_gemm_wmma<<<(gthreads + 255) / 256, 256, 0, stream>>>(
        agg0, Wl0, x, Wr0, b0, h0, M, IN_CH, HID, /*relu=*/1);

    // ---- layer 1 ----
    long long tot1 = (long long)E * (HID / 4);
    scatter_kernel4<<<(unsigned)((tot1 + 255) / 256), 256, 0, stream>>>(
        h0, src, dst, ew, agg1, tot1, 8 /* log2(256) */);
    long long sc1 = (long long)M * (HID / 4);
    scale_kernel4<<<(unsigned)((sc1 + 255) / 256), 256, 0, stream>>>(
        (float4*)agg1, deg, sc1, 6 /* log2(256/4) */);

    sage_gemm_wmma<<<(gthreads + 255) / 256, 256, 0, stream>>>(
        agg1, Wl1, h0, Wr1, b1, out, M, HID, HID, /*relu=*/0);
}